// Net_80109730005653
// MI455X (gfx1250) — compile-verified
//
#include <hip/hip_runtime.h>

// ---------------------------------------------------------------------------
// Fully fused GMT forward for MI455X (gfx1250, wave32).
// One workgroup (128 threads = 4 waves) per graph; all per-graph tensors in
// LDS; every GEMM runs on V_WMMA_F32_16X16X4_F32 (fp32 exact vs reference).
// ---------------------------------------------------------------------------

#define NPG      64    // nodes per graph
#define EPG      512   // edges per graph (NPG * DEG)
#define FIN      3
#define HGCN     32
#define CATD     96
#define HID      64
#define NK1      16    // pma1 seeds
#define HEADS    4
#define DH       16    // head dim
#define OUTP     32
#define NTHREADS 128
#define NWAVES   4

typedef __attribute__((ext_vector_type(2))) float v2f;
typedef __attribute__((ext_vector_type(8))) float v8f;

struct GmtParams {
    const float *x;
    const float *W1,*b1,*W2,*b2,*W3,*b3;          // conv1..3
    const float *Wp1,*bp1;                        // pool_lin1
    const float *S1,*Wq1,*bq1,*Wk1,*bk1,*Wv1,*bv1,*Wo1,*bo1;  // pma1
    const float *Wqs,*bqs,*Wks,*bks,*Wvs,*bvs,*Wos,*bos;      // sab
    const float *S2,*Wq2,*bq2,*Wk2,*bk2,*Wv2,*bv2,*Wo2,*bo2;  // pma2
    const float *Wp2,*bp2,*Wl1,*bl1,*Wl2,*bl2;    // pool_lin2, lin1, lin2
    const int   *edge;
    int          E;
    float       *out;
};

// --- generic WMMA f32 GEMM: C[M,N] = A[M,K] @ B[K,N] (+bias)(+relu) ---------
// A,B,C row-major; M,N multiples of 16, K multiple of 4; runs on all 4 waves.
// FLAGS: bit0 = add bias, bit1 = relu.
template<int FLAGS>
__device__ __forceinline__ void gemm_wmma(const float* A, int lda,
                                          const float* B, int ldb,
                                          float* C, int ldc,
                                          int M, int N, int K,
                                          const float* bias)
{
    const int lane = threadIdx.x & 31;
    const int m    = lane & 15;
    const int half = lane >> 4;
    const int tn   = N >> 4;
    const int ntiles = (M >> 4) * tn;
    for (int t = (int)(threadIdx.x >> 5); t < ntiles; t += NWAVES) {
        const int tr = (t / tn) << 4;
        const int tc = (t % tn) << 4;
        v8f acc = {0.f,0.f,0.f,0.f,0.f,0.f,0.f,0.f};
        for (int k = 0; k < K; k += 4) {
            const int ka = k + 2*half;           // A/B VGPRj <-> K = ka + j
            v2f a, b;
            a.x = A[(tr+m)*lda + ka];
            a.y = A[(tr+m)*lda + ka + 1];
            b.x = B[ ka     *ldb + tc + m];
            b.y = B[(ka + 1)*ldb + tc + m];
            acc = __builtin_amdgcn_wmma_f32_16x16x4_f32(false, a, false, b,
                                                        (short)0, acc, false, false);
        }
#pragma unroll
        for (int j = 0; j < 8; ++j) {            // C VGPRj <-> row j + 8*half
            const int row = tr + j + 8*half;
            const int col = tc + m;
            float v = acc[j];
            if (FLAGS & 1) v += bias[col];
            if (FLAGS & 2) v = fmaxf(v, 0.f);
            C[row*ldc + col] = v;
        }
    }
}

// --- attention scores: Sc[h][q][k] = sum_d Q[q][h*16+d] * K[k][h*16+d] ------
__device__ __forceinline__ void attn_scores(const float* Qm, const float* Km,
                                            int nk, float* Sc)
{
    const int lane = threadIdx.x & 31;
    const int m    = lane & 15;
    const int half = lane >> 4;
    const int ktiles = nk >> 4;
    for (int t = (int)(threadIdx.x >> 5); t < HEADS*ktiles; t += NWAVES) {
        const int h  = t / ktiles;
        const int kt = (t % ktiles) << 4;
        v8f acc = {0.f,0.f,0.f,0.f,0.f,0.f,0.f,0.f};
        for (int d = 0; d < DH; d += 4) {
            const int ka = d + 2*half;
            v2f a, b;
            a.x = Qm[m*HID + h*DH + ka];
            a.y = Qm[m*HID + h*DH + ka + 1];
            b.x = Km[(kt+m)*HID + h*DH + ka];       // B^T access (K stored [k][c])
            b.y = Km[(kt+m)*HID + h*DH + ka + 1];
            acc = __builtin_amdgcn_wmma_f32_16x16x4_f32(false, a, false, b,
                                                        (short)0, acc, false, false);
        }
#pragma unroll
        for (int j = 0; j < 8; ++j)
            Sc[(h*16 + j + 8*half)*64 + kt + m] = acc[j];
    }
}

// --- softmax over k (scale 1/sqrt(HID) = 0.125), probabilities in place -----
__device__ __forceinline__ void softmax_rows(float* Sc, int nk)
{
    for (int r = threadIdx.x; r < HEADS*16; r += NTHREADS) {
        float* row = Sc + r*64;
        float mx = -1e30f;
        for (int k = 0; k < nk; ++k) mx = fmaxf(mx, row[k]*0.125f);
        float s = 0.f;
        for (int k = 0; k < nk; ++k) { float e = expf(row[k]*0.125f - mx); row[k] = e; s += e; }
        const float inv = 1.f / s;
        for (int k = 0; k < nk; ++k) row[k] *= inv;
    }
}

// --- Om[q][h*16+d] = Qm[q][h*16+d] + sum_k P[h][q][k]*V[k][h*16+d];  wave=head
__device__ __forceinline__ void attn_av(const float* Sc, int nk, const float* Vm,
                                        const float* Qm, float* Om)
{
    const int lane = threadIdx.x & 31;
    const int m    = lane & 15;
    const int half = lane >> 4;
    const int h    = threadIdx.x >> 5;           // 4 waves = 4 heads
    const float* Ph = Sc + h*16*64;
    v8f acc = {0.f,0.f,0.f,0.f,0.f,0.f,0.f,0.f};
    for (int k = 0; k < nk; k += 4) {
        const int ka = k + 2*half;
        v2f a, b;
        a.x = Ph[m*64 + ka];
        a.y = Ph[m*64 + ka + 1];
        b.x = Vm[ ka     *HID + h*DH + m];
        b.y = Vm[(ka + 1)*HID + h*DH + m];
        acc = __builtin_amdgcn_wmma_f32_16x16x4_f32(false, a, false, b,
                                                    (short)0, acc, false, false);
    }
#pragma unroll
    for (int j = 0; j < 8; ++j) {
        const int row = j + 8*half;
        const int col = h*DH + m;
        Om[row*HID + col] = Qm[row*HID + col] + acc[j];
    }
}

__device__ __forceinline__ void mha(const float* Qm, const float* Km, const float* Vm,
                                    int nk, float* Sc, float* Om)
{
    attn_scores(Qm, Km, nk, Sc);   __syncthreads();
    softmax_rows(Sc, nk);          __syncthreads();
    attn_av(Sc, nk, Vm, Qm, Om);   __syncthreads();
}

// --- MAB tail: O += relu(O @ Wo + bo) ---------------------------------------
__device__ __forceinline__ void mab_fco(float* Om, float* Tmp,
                                        const float* Wo, const float* bo)
{
    gemm_wmma<3>(Om, HID, Wo, HID, Tmp, HID, NK1, HID, HID, bo);
    __syncthreads();
    for (int i = threadIdx.x; i < NK1*HID; i += NTHREADS) Om[i] += Tmp[i];
    __syncthreads();
}

// --- GCN aggregation helpers ------------------------------------------------
__device__ __forceinline__ void gcn_zero(float* outp, int ldo, int C)
{
    for (int i = threadIdx.x; i < NPG*C; i += NTHREADS)
        outp[(i/C)*ldo + (i % C)] = 0.f;
}
__device__ __forceinline__ void gcn_scatter(const float* lin, int ldl,
                                            float* outp, int ldo, int C,
                                            const int* sSrc, const int* sDst,
                                            const float* sCoef)
{
    for (int i = threadIdx.x; i < EPG*C; i += NTHREADS) {
        const int e = i / C, c = i - e*C;
        atomicAdd(&outp[sDst[e]*ldo + c], lin[sSrc[e]*ldl + c] * sCoef[e]);
    }
}
template<bool RELU>
__device__ __forceinline__ void gcn_finish(const float* lin, int ldl,
                                           float* outp, int ldo, int C,
                                           const float* dinv, const float* bias)
{
    for (int i = threadIdx.x; i < NPG*C; i += NTHREADS) {
        const int n = i / C, c = i - n*C;
        float v = outp[n*ldo + c] + lin[n*ldl + c]*dinv[n]*dinv[n] + bias[c];
        if (RELU) v = fmaxf(v, 0.f);
        outp[n*ldo + c] = v;
    }
}

// ---------------------------------------------------------------------------
__global__ __launch_bounds__(NTHREADS, 1)
void gmt_fused_kernel(GmtParams p)
{
    const int g   = blockIdx.x;
    const int tid = threadIdx.x;

    __shared__ float sCat[NPG*CATD];     // [64,96] = relu(conv1)|relu(conv2)|relu(conv3)
    __shared__ float sH  [NPG*HID];      // [64,64] pool_lin1 output
    __shared__ float sTS [HEADS*16*64];  // aliased: GEMM/GCN scratch <-> attn scores
    __shared__ float sK  [NPG*HID];
    __shared__ float sV  [NPG*HID];
    __shared__ float sQ  [NK1*HID];      // also stages x[64,3] before pma1
    __shared__ float sO  [NK1*HID];
    __shared__ float sO2 [NK1*HID];
    __shared__ int   sSrc[EPG];
    __shared__ int   sDst[EPG];
    __shared__ float sCoef[EPG];
    __shared__ float sDinv[NPG];
    __shared__ float sSmall[128];

    // ---- edges, degrees, symmetric-norm coefficients -----------------------
    if (tid < NPG) sDinv[tid] = 0.f;
    float* sX = sQ;                              // stage x here for conv1
    for (int i = tid; i < NPG*FIN; i += NTHREADS) sX[i] = p.x[(long)g*NPG*FIN + i];
    __syncthreads();
    for (int e = tid; e < EPG; e += NTHREADS) {
        const int s = p.edge[(long)g*EPG + e]              - g*NPG;
        const int d = p.edge[(long)p.E + (long)g*EPG + e]  - g*NPG;
        sSrc[e] = s; sDst[e] = d;
        atomicAdd(&sDinv[d], 1.f);
    }
    __syncthreads();
    if (tid < NPG) sDinv[tid] = rsqrtf(sDinv[tid] + 1.f);  // +1 self loop
    __syncthreads();
    for (int e = tid; e < EPG; e += NTHREADS) sCoef[e] = sDinv[sSrc[e]] * sDinv[sDst[e]];
    __syncthreads();

    // ---- conv1: [64,3]@[3,32], aggregate, relu -> sCat[:,0:32] -------------
    for (int i = tid; i < NPG*HGCN; i += NTHREADS) {
        const int n = i >> 5, c = i & 31;
        sTS[n*64 + c] = sX[n*FIN+0]*p.W1[0*HGCN+c]
                      + sX[n*FIN+1]*p.W1[1*HGCN+c]
                      + sX[n*FIN+2]*p.W1[2*HGCN+c];
    }
    __syncthreads();
    gcn_zero(sCat + 0, CATD, HGCN);                               __syncthreads();
    gcn_scatter(sTS, 64, sCat + 0, CATD, HGCN, sSrc, sDst, sCoef);__syncthreads();
    gcn_finish<true>(sTS, 64, sCat + 0, CATD, HGCN, sDinv, p.b1); __syncthreads();

    // ---- conv2 -> sCat[:,32:64] --------------------------------------------
    gemm_wmma<0>(sCat + 0, CATD, p.W2, HGCN, sTS, 64, NPG, HGCN, HGCN, nullptr);
    __syncthreads();
    gcn_zero(sCat + HGCN, CATD, HGCN);                                __syncthreads();
    gcn_scatter(sTS, 64, sCat + HGCN, CATD, HGCN, sSrc, sDst, sCoef); __syncthreads();
    gcn_finish<true>(sTS, 64, sCat + HGCN, CATD, HGCN, sDinv, p.b2);  __syncthreads();

    // ---- conv3 -> sCat[:,64:96] --------------------------------------------
    gemm_wmma<0>(sCat + HGCN, CATD, p.W3, HGCN, sTS, 64, NPG, HGCN, HGCN, nullptr);
    __syncthreads();
    gcn_zero(sCat + 2*HGCN, CATD, HGCN);                                __syncthreads();
    gcn_scatter(sTS, 64, sCat + 2*HGCN, CATD, HGCN, sSrc, sDst, sCoef);__syncthreads();
    gcn_finish<true>(sTS, 64, sCat + 2*HGCN, CATD, HGCN, sDinv, p.b3); __syncthreads();

    // ---- pool_lin1: h = cat @ Wp1 + b  [64,96]@[96,64] ---------------------
    gemm_wmma<1>(sCat, CATD, p.Wp1, HID, sH, HID, NPG, HID, CATD, p.bp1);
    __syncthreads();

    // ---- pma1: K = gcn(layer_k, h), V = gcn(layer_v, h) --------------------
    gemm_wmma<0>(sH, HID, p.Wk1, HID, sTS, 64, NPG, HID, HID, nullptr); __syncthreads();
    gcn_zero(sK, HID, HID);                                   __syncthreads();
    gcn_scatter(sTS, 64, sK, HID, HID, sSrc, sDst, sCoef);    __syncthreads();
    gcn_finish<false>(sTS, 64, sK, HID, HID, sDinv, p.bk1);   __syncthreads();

    gemm_wmma<0>(sH, HID, p.Wv1, HID, sTS, 64, NPG, HID, HID, nullptr); __syncthreads();
    gcn_zero(sV, HID, HID);                                   __syncthreads();
    gcn_scatter(sTS, 64, sV, HID, HID, sSrc, sDst, sCoef);    __syncthreads();
    gcn_finish<false>(sTS, 64, sV, HID, HID, sDinv, p.bv1);   __syncthreads();

    // ---- pma1: Q = S1 @ Wq1 + b  (x staging in sQ is dead now) -------------
    gemm_wmma<1>(p.S1, HID, p.Wq1, HID, sQ, HID, NK1, HID, HID, p.bq1);
    __syncthreads();

    mha(sQ, sK, sV, NPG, sTS, sO);         // sO = Q + softmax(QK^T/8)V
    mab_fco(sO, sO2, p.Wo1, p.bo1);        // sO += relu(sO@Wo1+bo1)

    // ---- SAB: Q=K=V projections from sO ------------------------------------
    gemm_wmma<1>(sO, HID, p.Wqs, HID, sQ, HID, NK1, HID, HID, p.bqs);
    gemm_wmma<1>(sO, HID, p.Wks, HID, sK, HID, NK1, HID, HID, p.bks);
    gemm_wmma<1>(sO, HID, p.Wvs, HID, sV, HID, NK1, HID, HID, p.bvs);
    __syncthreads();
    mha(sQ, sK, sV, NK1, sTS, sO);
    mab_fco(sO, sO2, p.Wos, p.bos);

    // ---- pma2 (1 seed) ------------------------------------------------------
    gemm_wmma<1>(sO, HID, p.Wk2, HID, sK, HID, NK1, HID, HID, p.bk2);
    gemm_wmma<1>(sO, HID, p.Wv2, HID, sV, HID, NK1, HID, HID, p.bv2);
    __syncthreads();
    if (tid < HID) {                       // Q2 = S2 @ Wq2 + b (graph-invariant)
        float a = p.bq2[tid];
        for (int j = 0; j < HID; ++j) a += p.S2[j] * p.Wq2[j*HID + tid];
        sQ[tid] = a;
    }
    __syncthreads();
    if (tid < HEADS*DH) {                  // scores[h][k], nq = 1
        const int h = tid >> 4, kk = tid & 15;
        float s = 0.f;
        for (int d = 0; d < DH; ++d) s += sQ[h*DH + d] * sK[kk*HID + h*DH + d];
        sTS[tid] = s * 0.125f;
    }
    __syncthreads();
    if (tid < HEADS) {                     // softmax over 16 keys per head
        float mx = -1e30f;
        for (int k = 0; k < 16; ++k) mx = fmaxf(mx, sTS[tid*16 + k]);
        float s = 0.f;
        for (int k = 0; k < 16; ++k) { float e = expf(sTS[tid*16 + k] - mx); sTS[tid*16 + k] = e; s += e; }
        const float inv = 1.f / s;
        for (int k = 0; k < 16; ++k) sTS[tid*16 + k] *= inv;
    }
    __syncthreads();
    if (tid < HID) {                       // out = Q2 + A@V
        const int h = tid >> 4;
        float a = 0.f;
        for (int k = 0; k < 16; ++k) a += sTS[h*16 + k] * sV[k*HID + tid];
        sO[tid] = sQ[tid] + a;
    }
    __syncthreads();
    if (tid < HID) {                       // fc_o
        float t = p.bo2[tid];
        for (int j = 0; j < HID; ++j) t += sO[j] * p.Wo2[j*HID + tid];
        sO2[tid] = t;
    }
    __syncthreads();
    if (tid < HID) sSmall[tid] = sO[tid] + fmaxf(sO2[tid], 0.f);
    __syncthreads();

    // ---- head MLP -----------------------------------------------------------
    if (tid < OUTP) {                      // pool_lin2: 64 -> 32
        float v = p.bp2[tid];
        for (int j = 0; j < HID; ++j) v += sSmall[j] * p.Wp2[j*OUTP + tid];
        sSmall[64 + tid] = v;
    }
    __syncthreads();
    if (tid < 16) {                        // lin1: 32 -> 16, relu
        float v = p.bl1[tid];
        for (int j = 0; j < OUTP; ++j) v += sSmall[64 + j] * p.Wl1[j*16 + tid];
        sSmall[96 + tid] = fmaxf(v, 0.f);
    }
    __syncthreads();
    if (tid < 2) {                         // lin2: 16 -> 2
        float v = p.bl2[tid];
        for (int j = 0; j < 16; ++j) v += sSmall[96 + j] * p.Wl2[j*2 + tid];
        sSmall[112 + tid] = v;
    }
    __syncthreads();
    if (tid == 0) {                        // log_softmax
        const float a = sSmall[112], b = sSmall[113];
        const float mx  = fmaxf(a, b);
        const float lse = mx + logf(expf(a - mx) + expf(b - mx));
        p.out[(long)g*2 + 0] = a - lse;
        p.out[(long)g*2 + 1] = b - lse;
    }
}

// ---------------------------------------------------------------------------
extern "C" void kernel_launch(void* const* d_in, const int* in_sizes, int n_in,
                              void* d_out, int out_size, void* d_ws, size_t ws_size,
                              hipStream_t stream)
{
    (void)n_in; (void)out_size; (void)d_ws; (void)ws_size;
    GmtParams p;
    p.x   = (const float*)d_in[0];
    p.W1  = (const float*)d_in[1];  p.b1  = (const float*)d_in[2];
    p.W2  = (const float*)d_in[3];  p.b2  = (const float*)d_in[4];
    p.W3  = (const float*)d_in[5];  p.b3  = (const float*)d_in[6];
    p.Wp1 = (const float*)d_in[7];  p.bp1 = (const float*)d_in[8];
    p.S1  = (const float*)d_in[9];
    p.Wq1 = (const float*)d_in[10]; p.bq1 = (const float*)d_in[11];
    p.Wk1 = (const float*)d_in[12]; p.bk1 = (const float*)d_in[13];
    p.Wv1 = (const float*)d_in[14]; p.bv1 = (const float*)d_in[15];
    p.Wo1 = (const float*)d_in[16]; p.bo1 = (const float*)d_in[17];
    p.Wqs = (const float*)d_in[18]; p.bqs = (const float*)d_in[19];
    p.Wks = (const float*)d_in[20]; p.bks = (const float*)d_in[21];
    p.Wvs = (const float*)d_in[22]; p.bvs = (const float*)d_in[23];
    p.Wos = (const float*)d_in[24]; p.bos = (const float*)d_in[25];
    p.S2  = (const float*)d_in[26];
    p.Wq2 = (const float*)d_in[27]; p.bq2 = (const float*)d_in[28];
    p.Wk2 = (const float*)d_in[29]; p.bk2 = (const float*)d_in[30];
    p.Wv2 = (const float*)d_in[31]; p.bv2 = (const float*)d_in[32];
    p.Wo2 = (const float*)d_in[33]; p.bo2 = (const float*)d_in[34];
    p.Wp2 = (const float*)d_in[35]; p.bp2 = (const float*)d_in[36];
    p.Wl1 = (const float*)d_in[37]; p.bl1 = (const float*)d_in[38];
    p.Wl2 = (const float*)d_in[39]; p.bl2 = (const float*)d_in[40];
    p.edge = (const int*)d_in[41];
    p.E    = in_sizes[41] / 2;
    p.out  = (float*)d_out;

    const int N = in_sizes[0] / FIN;     // 131072 nodes
    const int B = N / NPG;               // 2048 graphs
    gmt_fused_kernel<<<dim3(B), dim3(NTHREADS), 0, stream>>>(p);
}